// ParallelOPTLearnedPositionalEmbedding_11295763988602
// MI455X (gfx1250) — compile-verified
//
#include <hip/hip_runtime.h>

// ParallelOPTLearnedPositionalEmbedding for MI455X (gfx1250).
// Pure gather + axpy: out[p,b,s,:] = weight[pos[b,s],:] + 0.01 * mu[p,pos[b,s],:]
// Memory-bound (1 GiB NT output stream, mu/weight L2-resident). No matrix math.

#define OFFSET_C   2
#define EPS_C      0.01f
#define NUM_EMB_C  2050        // 2048 + OFFSET
#define EMBED_C    2048
#define NPAR_C     8
#define BATCH_C    8
#define SEQ_C      2048

typedef __attribute__((ext_vector_type(4))) float v4f;

// ---------------------------------------------------------------------------
// Kernel 1: positions[b,s] = cumsum(mask[b,:])[s] * mask[b,s] - 1 + OFFSET
// One block per batch row. Each lane scans 8 contiguous elements in registers;
// lane partials are scanned through LDS.
// ---------------------------------------------------------------------------
__global__ __launch_bounds__(256)
void opt_positions_kernel(const int* __restrict__ mask,
                          int* __restrict__ pos_out)
{
    __shared__ int partial[256];
    const int b = blockIdx.x;
    const int t = threadIdx.x;
    const int* row = mask + (size_t)b * SEQ_C;

    int m[8];
    int run = 0;
#pragma unroll
    for (int i = 0; i < 8; ++i) {
        m[i] = row[t * 8 + i];
        run += m[i];
    }
    partial[t] = run;
    __syncthreads();

    if (t == 0) {
        int acc = 0;
#pragma unroll 4
        for (int i = 0; i < 256; ++i) {
            int v = partial[i];
            partial[i] = acc;   // exclusive prefix of lane partials
            acc += v;
        }
    }
    __syncthreads();

    int cum = partial[t];
#pragma unroll
    for (int i = 0; i < 8; ++i) {
        cum += m[i];                                  // inclusive cumsum at element
        int p = cum * m[i] - 1 + OFFSET_C;            // masked position + OFFSET
        pos_out[(size_t)b * SEQ_C + t * 8 + i] = p;
    }
}

// ---------------------------------------------------------------------------
// Kernel 2: one block per (b,s). Weight row held in registers (2 x float4 per
// lane), loop over the 8 parallel mu planes: B128 loads, NT B128 stores so the
// 1 GiB output doesn't evict the L2-resident 134 MB mu tensor.
// ---------------------------------------------------------------------------
__global__ __launch_bounds__(256)
void opt_embed_axpy_kernel(const float* __restrict__ weight,
                           const float* __restrict__ mu,
                           const int*   __restrict__ pos,
                           const int*   __restrict__ pkv_ptr,
                           float*       __restrict__ out)
{
    const int pkv     = pkv_ptr[0];
    const int seq_out = SEQ_C - pkv;

    const int bs = blockIdx.x;            // 0 .. BATCH*SEQ-1
    const int b  = bs / SEQ_C;
    const int s  = bs % SEQ_C;
    if (s >= seq_out) return;

    const int t = threadIdx.x;

    // Block-uniform gather index (address depends only on blockIdx -> SMEM path)
    const int p = pos[(size_t)b * SEQ_C + (s + pkv)];

    const v4f* wrow = (const v4f*)(weight + (size_t)p * EMBED_C);
    const v4f w0 = wrow[t];
    const v4f w1 = wrow[t + 256];

    const size_t mu_plane  = (size_t)NUM_EMB_C * EMBED_C;        // stride between p-planes
    const float* mu_base   = mu + (size_t)p * EMBED_C;           // gathered row, plane 0
    const size_t out_plane = (size_t)BATCH_C * seq_out * EMBED_C;
    float*       out_base  = out + ((size_t)b * seq_out + s) * EMBED_C;

#pragma unroll
    for (int pp = 0; pp < NPAR_C; ++pp) {
        const v4f* mrow = (const v4f*)(mu_base + (size_t)pp * mu_plane);

        if (pp + 1 < NPAR_C) {
            // gfx1250 global_prefetch_b8: pull next plane's gathered row toward L2/L0
            __builtin_prefetch((const void*)(mu_base + (size_t)(pp + 1) * mu_plane
                                             + (size_t)t * 4), 0, 1);
        }

        const v4f m0 = mrow[t];
        const v4f m1 = mrow[t + 256];

        const v4f o0 = w0 + EPS_C * m0;
        const v4f o1 = w1 + EPS_C * m1;

        v4f* orow = (v4f*)(out_base + (size_t)pp * out_plane);
        __builtin_nontemporal_store(o0, orow + t);
        __builtin_nontemporal_store(o1, orow + t + 256);
    }
}

// ---------------------------------------------------------------------------
// Launch
//   d_in[0] attention_mask (int, 8*2048)
//   d_in[1] past_key_values_length (int, 1)
//   d_in[2] weight (float, 2050*2048)
//   d_in[3] mu (float, 8*2050*2048)
//   d_out   float, 8*8*2048*2048
//   d_ws    >= 8*2048*sizeof(int) used for positions
// ---------------------------------------------------------------------------
extern "C" void kernel_launch(void* const* d_in, const int* in_sizes, int n_in,
                              void* d_out, int out_size, void* d_ws, size_t ws_size,
                              hipStream_t stream)
{
    const int*   mask    = (const int*)d_in[0];
    const int*   pkv     = (const int*)d_in[1];
    const float* weight  = (const float*)d_in[2];
    const float* mu      = (const float*)d_in[3];
    float*       out     = (float*)d_out;
    int*         pos_ws  = (int*)d_ws;   // 64 KB

    (void)in_sizes; (void)n_in; (void)out_size; (void)ws_size;

    opt_positions_kernel<<<BATCH_C, 256, 0, stream>>>(mask, pos_ws);

    opt_embed_axpy_kernel<<<BATCH_C * SEQ_C, 256, 0, stream>>>(
        weight, mu, pos_ws, pkv, out);
}